// ParallelConv1D_40690520162744
// MI455X (gfx1250) — compile-verified
//
#include <hip/hip_runtime.h>

// ParallelConv1D: per-channel conv1d(K=5,'same') + maxpool2, via
// V_WMMA_F32_16X16X4_F32 (full f32 precision; kernel is store-BW bound).

typedef __attribute__((ext_vector_type(2))) float v2f;
typedef __attribute__((ext_vector_type(8))) float v8f;

#define C_DIM 16
#define L_DIM 1024
#define K_DIM 5
#define F_DIM 16
#define LP_DIM 512   // L/2 pooled

__global__ __launch_bounds__(256) void conv_pool_wmma_kernel(
    const float* __restrict__ x,     // (B*T*C, L)
    const float* __restrict__ W,     // (C, K, F)
    const float* __restrict__ bias,  // (C, F)
    float* __restrict__ out)         // (B*T*C, Lp, F)
{
  const int row  = blockIdx.x;            // bt*C + c
  const int c    = row & (C_DIM - 1);
  const int lane = threadIdx.x & 31;
  const int wave = threadIdx.x >> 5;      // 0..7
  const int half = lane >> 4;             // 0 or 1
  const int fn   = lane & 15;             // N index = filter
  const int m    = lane & 15;             // A-matrix row (M)
  const int kb   = half * 2;              // K-pair base for this lane half

  const float* __restrict__ xrow = x + (size_t)row * L_DIM;
  const float* __restrict__ Wc   = W + c * (K_DIM * F_DIM);
  float* __restrict__ orow       = out + (size_t)row * (LP_DIM * F_DIM);

  // B-matrix (4x16) for taps 0..3: VGPR v, lanes0-15 -> K=v, lanes16-31 -> K=v+2
  v2f b1;
  b1.x = Wc[(kb + 0) * F_DIM + fn];
  b1.y = Wc[(kb + 1) * F_DIM + fn];
  // B-matrix for tap 4 in K-slot 0 only
  v2f b2;
  b2.x = (half == 0) ? Wc[4 * F_DIM + fn] : 0.0f;
  b2.y = 0.0f;

  const float bv = bias[c * F_DIM + fn];

  // 8 waves x 8 tiles of 16 positions = 1024 positions per row
  for (int t = 0; t < 8; ++t) {
    const int l0 = (wave * 8 + t) * 16;

    // A1[m][k] = x[l0 + m - 2 + k], k in {kb, kb+1}; A2[m][0] = x[l0 + m + 2]
    const int i0 = l0 + m - 2 + kb;
    const int i1 = i0 + 1;
    const int i2 = l0 + m + 2;

    // Clamped loads + select-zero (v_cndmask) -> no EXEC divergence at WMMA
    int c0 = i0 < 0 ? 0 : (i0 > L_DIM - 1 ? L_DIM - 1 : i0);
    int c1 = i1 < 0 ? 0 : (i1 > L_DIM - 1 ? L_DIM - 1 : i1);
    int c2 = i2 > L_DIM - 1 ? L_DIM - 1 : i2;
    float a0 = xrow[c0];
    float a1 = xrow[c1];
    float a2 = xrow[c2];
    a0 = ((unsigned)i0 < (unsigned)L_DIM) ? a0 : 0.0f;
    a1 = ((unsigned)i1 < (unsigned)L_DIM) ? a1 : 0.0f;
    a2 = (half == 0 && i2 < L_DIM) ? a2 : 0.0f;

    v2f A1; A1.x = a0; A1.y = a1;
    v2f A2; A2.x = a2; A2.y = 0.0f;

    v8f acc = {};
    // taps {4} then taps {0..3}: D = A1*B1 + (A2*B2 + 0)
    acc = __builtin_amdgcn_wmma_f32_16x16x4_f32(
        false, A2, false, b2, (short)0, acc, false, false);
    acc = __builtin_amdgcn_wmma_f32_16x16x4_f32(
        false, A1, false, b1, (short)0, acc, false, false);

    // acc[v] = y[l0 + v + 8*half][fn]; pool adjacent rows, add bias, store.
    const int lp0 = (l0 >> 1) + half * 4;
#pragma unroll
    for (int p = 0; p < 4; ++p) {
      float y0 = acc[2 * p + 0] + bv;
      float y1 = acc[2 * p + 1] + bv;
      float pv = fmaxf(y0, y1);
      orow[(size_t)(lp0 + p) * F_DIM + fn] = pv;
    }
  }
}

extern "C" void kernel_launch(void* const* d_in, const int* in_sizes, int n_in,
                              void* d_out, int out_size, void* d_ws, size_t ws_size,
                              hipStream_t stream) {
  (void)in_sizes; (void)n_in; (void)out_size; (void)d_ws; (void)ws_size;
  const float* x    = (const float*)d_in[0];  // (16,32,16,1024)
  const float* W    = (const float*)d_in[1];  // (16,5,16)
  const float* bias = (const float*)d_in[2];  // (16,16)
  float* out        = (float*)d_out;          // (16,32,16,512,16)

  const int rows = 16 * 32 * 16;              // B*T*C = 8192 blocks
  conv_pool_wmma_kernel<<<rows, 256, 0, stream>>>(x, W, bias, out);
}